// KalmanVAE_73512660238710
// MI455X (gfx1250) — compile-verified
//
#include <hip/hip_runtime.h>
#include <cstddef>

#define BB 512
#define TT 128
#define OBSD 32
#define LATD 16
#define KK 3
#define HIDD 50
#define Q_SCALE 0.08f
#define R_SCALE 0.03f
#define SIGMA1_SCALE 20.0f

typedef float v2f __attribute__((ext_vector_type(2)));
typedef float v8f __attribute__((ext_vector_type(8)));

__device__ inline v8f zero8() {
    v8f z = {0.f, 0.f, 0.f, 0.f, 0.f, 0.f, 0.f, 0.f};
    return z;
}

// D(16x16) += opA(A) * (bscale * opB(B)) over K = KDIM (multiple of 4), fp32 WMMA.
// Wave32 layouts (CDNA5 ISA 7.12.2):
//   A 16x4 frag: lanes 0-15 row m hold K={k0,k0+1}; lanes 16-31 row m hold K={k0+2,k0+3}
//   B 4x16 frag: mirrored (rows striped across lanes within a VGPR)
//   C/D 16x16:   VGPR v holds row (v + 8*(lane>=16)), col = lane&15
template <int KDIM, bool TA, bool TB>
__device__ inline v8f wmma_mm(const float* A, int lda, const float* B, int ldb,
                              float bscale, v8f acc) {
    int lane = threadIdx.x & 31;
    int half = lane >> 4;
    int m = lane & 15;
#pragma unroll
    for (int k0 = 0; k0 < KDIM; k0 += 4) {
        int ka = k0 + 2 * half;
        v2f a, b;
        if (!TA) { a.x = A[m * lda + ka];     a.y = A[m * lda + ka + 1]; }
        else     { a.x = A[ka * lda + m];     a.y = A[(ka + 1) * lda + m]; }
        if (!TB) { b.x = B[ka * ldb + m];     b.y = B[(ka + 1) * ldb + m]; }
        else     { b.x = B[m * ldb + ka];     b.y = B[m * ldb + ka + 1]; }
        b.x *= bscale; b.y *= bscale;
        acc = __builtin_amdgcn_wmma_f32_16x16x4_f32(false, a, false, b,
                                                    (short)0, acc, false, false);
    }
    return acc;
}

__device__ inline void store_c(float* C, int ldc, v8f acc) {
    int lane = threadIdx.x & 31;
    int half = lane >> 4;
    int n = lane & 15;
#pragma unroll
    for (int v = 0; v < 8; ++v) C[(v + 8 * half) * ldc + n] = acc[v];
}

__device__ inline v8f load_c(const float* C, int ldc) {
    int lane = threadIdx.x & 31;
    int half = lane >> 4;
    int n = lane & 15;
    v8f acc;
#pragma unroll
    for (int v = 0; v < 8; ++v) acc[v] = C[(v + 8 * half) * ldc + n];
    return acc;
}

// In-place Cholesky (lower) of n x n SPD matrix in LDS, one lane per row.
__device__ inline void chol_n(float* S, int lds_, int n) {
    int lane = threadIdx.x;
    for (int k = 0; k < n; ++k) {
        __syncthreads();
        float dk = sqrtf(S[k * lds_ + k]);
        float inv = 1.0f / dk;
        __syncthreads();
        if (lane == k) S[k * lds_ + k] = dk;
        if (lane < n && lane > k) S[lane * lds_ + k] *= inv;
        __syncthreads();
        if (lane < n && lane > k) {
            float lik = S[lane * lds_ + k];
            for (int j = k + 1; j <= lane; ++j)
                S[lane * lds_ + j] -= lik * S[j * lds_ + k];
        }
        __syncthreads();
    }
}

// Solve (L L^T) X = Bm for X (n x 16), L lower-tri in LDS; one lane per RHS column.
__device__ inline void chol_solve(const float* L, int ldl, const float* Bm, int ldb,
                                  float* X, int ldx, int n) {
    int lane = threadIdx.x;
    if (lane < 16) {
        int c = lane;
        for (int i = 0; i < n; ++i) {           // forward: L z = B
            float s = Bm[i * ldb + c];
            for (int j = 0; j < i; ++j) s -= L[i * ldl + j] * X[j * ldx + c];
            X[i * ldx + c] = s / L[i * ldl + i];
        }
        for (int i = n - 1; i >= 0; --i) {      // backward: L^T x = z
            float s = X[i * ldx + c];
            for (int j = i + 1; j < n; ++j) s -= L[j * ldl + i] * X[j * ldx + c];
            X[i * ldx + c] = s / L[i * ldl + i];
        }
    }
    __syncthreads();
}

// ---------------- Kernel 1: mixture weights (MLP + softmax) ----------------
__global__ __launch_bounds__(256) void kweights(const float* __restrict__ obs,
                                                const float* __restrict__ start_code,
                                                const float* __restrict__ W1,
                                                const float* __restrict__ b1,
                                                const float* __restrict__ W2,
                                                const float* __restrict__ b2,
                                                float* __restrict__ w_out) {
    __shared__ float sW1[OBSD * HIDD];
    __shared__ float sb1[HIDD];
    __shared__ float sW2[HIDD * KK];
    __shared__ float sb2[KK];
    __shared__ float sstart[OBSD];
    int tid = threadIdx.x;
    for (int i = tid; i < OBSD * HIDD; i += 256) sW1[i] = W1[i];
    for (int i = tid; i < HIDD; i += 256) sb1[i] = b1[i];
    for (int i = tid; i < HIDD * KK; i += 256) sW2[i] = W2[i];
    if (tid < KK) sb2[tid] = b2[tid];
    if (tid < OBSD) sstart[tid] = start_code[tid];
    __syncthreads();

    int idx = blockIdx.x * 256 + tid;  // b*T + t
    if (idx >= BB * TT) return;
    int t = idx % TT;

    float x[OBSD];
    if (t == 0) {
        for (int i = 0; i < OBSD; ++i) x[i] = sstart[i];
    } else {
        const float* p = obs + (size_t)(idx - 1) * OBSD;  // obs[b][t-1]
        for (int i = 0; i < OBSD; ++i) x[i] = p[i];
    }
    float l0 = sb2[0], l1 = sb2[1], l2 = sb2[2];
    for (int h = 0; h < HIDD; ++h) {
        float s = sb1[h];
        for (int i = 0; i < OBSD; ++i) s += x[i] * sW1[i * HIDD + h];
        s = fmaxf(s, 0.f);
        l0 += s * sW2[h * KK + 0];
        l1 += s * sW2[h * KK + 1];
        l2 += s * sW2[h * KK + 2];
    }
    float mx = fmaxf(l0, fmaxf(l1, l2));
    float e0 = __expf(l0 - mx), e1 = __expf(l1 - mx), e2 = __expf(l2 - mx);
    float inv = 1.f / (e0 + e1 + e2);
    w_out[idx * 3 + 0] = e0 * inv;
    w_out[idx * 3 + 1] = e1 * inv;
    w_out[idx * 3 + 2] = e2 * inv;
}

// ---------------- Kernel 2: forward Kalman filter (1 wave per batch) ----------------
__global__ __launch_bounds__(32) void kfilter(const float* __restrict__ obs,
                                              const float* __restrict__ Abase_g,
                                              const float* __restrict__ Cbase_g,
                                              const float* __restrict__ w_g,
                                              float* __restrict__ mu_pred,
                                              float* __restrict__ Sig_pred,
                                              float* __restrict__ out_mu,
                                              float* __restrict__ out_Sig) {
    int b = blockIdx.x;
    int lane = threadIdx.x;

    __shared__ float Ab[KK * 256];
    __shared__ float Cb[KK * 512];
    __shared__ float Sig[16 * 17], T1[16 * 17], SigZ[16 * 17], Im[16 * 17], An[16 * 17];
    __shared__ float Ct[32 * 17], CS[32 * 17], KgT[32 * 17];
    __shared__ float S[32 * 33];
    __shared__ float mu[16], muz[16], rvec[32], wsh[6];

    for (int i = lane; i < KK * 256; i += 32) Ab[i] = Abase_g[i];
    for (int i = lane; i < KK * 512; i += 32) Cb[i] = Cbase_g[i];
    for (int i = lane; i < 16 * 17; i += 32) Sig[i] = 0.f;
    __syncthreads();
    if (lane < 16) { Sig[lane * 17 + lane] = SIGMA1_SCALE; mu[lane] = 0.f; }
    __syncthreads();

    for (int t = 0; t < TT; ++t) {
        if (lane < 3) {
            wsh[lane] = w_g[(b * TT + t) * 3 + lane];
            wsh[3 + lane] = (t + 1 < TT) ? w_g[(b * TT + t + 1) * 3 + lane] : 0.f;
        }
        __syncthreads();
        float w0 = wsh[0], w1 = wsh[1], w2 = wsh[2];
        float v0 = wsh[3], v1 = wsh[4], v2 = wsh[5];

        // C_t = sum_k w_k C[k]   (32x16)
        for (int i = lane; i < 512; i += 32) {
            int rr = i >> 4, cc = i & 15;
            Ct[rr * 17 + cc] = w0 * Cb[i] + w1 * Cb[512 + i] + w2 * Cb[1024 + i];
        }
        // A_next = sum_k w_{t+1,k} A[k], identity at t = T-1
        for (int i = lane; i < 256; i += 32) {
            int rr = i >> 4, cc = i & 15;
            float a = v0 * Ab[i] + v1 * Ab[256 + i] + v2 * Ab[512 + i];
            if (t + 1 >= TT) a = (rr == cc) ? 1.f : 0.f;
            An[rr * 17 + cc] = a;
        }
        // record predicted (prior) state
        if (lane < 16) mu_pred[(size_t)(b * TT + t) * 16 + lane] = mu[lane];
        for (int i = lane; i < 256; i += 32)
            Sig_pred[(size_t)(b * TT + t) * 256 + i] = Sig[(i >> 4) * 17 + (i & 15)];
        __syncthreads();

        // CS = C_t @ Sig  (two 16-row tiles)
        store_c(&CS[0],       17, wmma_mm<16, false, false>(&Ct[0],       17, Sig, 17, 1.f, zero8()));
        store_c(&CS[16 * 17], 17, wmma_mm<16, false, false>(&Ct[16 * 17], 17, Sig, 17, 1.f, zero8()));
        __syncthreads();

        // S = CS @ C_t^T + R   (32x32 in 4 tiles)
        for (int bi = 0; bi < 2; ++bi)
            for (int bj = 0; bj < 2; ++bj) {
                v8f acc = wmma_mm<16, false, true>(&CS[bi * 16 * 17], 17,
                                                   &Ct[bj * 16 * 17], 17, 1.f, zero8());
                store_c(&S[bi * 16 * 33 + bj * 16], 33, acc);
            }
        __syncthreads();
        S[lane * 33 + lane] += R_SCALE;
        __syncthreads();

        chol_n(S, 33, 32);
        // Kg^T = S^{-1} (C Sig)   (Sig symmetric => Sig C^T = CS^T)
        chol_solve(S, 33, CS, 17, KgT, 17, 32);

        // residual r = y - C_t mu
        {
            float yv = obs[((size_t)b * TT + t) * OBSD + lane];
            float s = yv;
            for (int j = 0; j < 16; ++j) s -= Ct[lane * 17 + j] * mu[j];
            rvec[lane] = s;
        }
        __syncthreads();
        if (lane < 16) {
            float s = mu[lane];
            for (int k = 0; k < 32; ++k) s += KgT[k * 17 + lane] * rvec[k];
            muz[lane] = s;
        }
        __syncthreads();

        // I- = I - Kg C_t   (Kg = KgT^T)
        store_c(Im, 17, wmma_mm<32, true, false>(KgT, 17, Ct, 17, 1.f, zero8()));
        __syncthreads();
        for (int i = lane; i < 256; i += 32) {
            int rr = i >> 4, cc = i & 15;
            Im[rr * 17 + cc] = ((rr == cc) ? 1.f : 0.f) - Im[rr * 17 + cc];
        }
        __syncthreads();

        // Sig_z = Kg (R Kg^T) + (I- Sig) I-^T  : chained accumulators
        v8f sz = wmma_mm<32, true, false>(KgT, 17, KgT, 17, R_SCALE, zero8());
        store_c(T1, 17, wmma_mm<16, false, false>(Im, 17, Sig, 17, 1.f, zero8()));
        __syncthreads();
        sz = wmma_mm<16, false, true>(T1, 17, Im, 17, 1.f, sz);
        store_c(SigZ, 17, sz);
        __syncthreads();

        // filtered results straight into d_out (smoother overwrites t < T-1 in place)
        if (lane < 16) out_mu[((size_t)t * BB + b) * 16 + lane] = muz[lane];
        for (int i = lane; i < 256; i += 32)
            out_Sig[((size_t)t * BB + b) * 256 + i] = SigZ[(i >> 4) * 17 + (i & 15)];

        // Sig = A_n Sig_z A_n^T + Q ; mu = A_n mu_z
        store_c(T1, 17, wmma_mm<16, false, false>(An, 17, SigZ, 17, 1.f, zero8()));
        __syncthreads();
        store_c(Sig, 17, wmma_mm<16, false, true>(T1, 17, An, 17, 1.f, zero8()));
        __syncthreads();
        if (lane < 16) {
            Sig[lane * 17 + lane] += Q_SCALE;
            float s = 0.f;
            for (int j = 0; j < 16; ++j) s += An[lane * 17 + j] * muz[j];
            mu[lane] = s;
        }
        __syncthreads();
    }
}

// ---------------- Kernel 3: backward RTS smoother (1 wave per batch) ----------------
__global__ __launch_bounds__(32) void ksmooth(const float* __restrict__ Abase_g,
                                              const float* __restrict__ w_g,
                                              const float* __restrict__ mu_pred,
                                              const float* __restrict__ Sig_pred,
                                              float* __restrict__ out_mu,
                                              float* __restrict__ out_Sig) {
    int b = blockIdx.x;
    int lane = threadIdx.x;

    __shared__ float Ab[KK * 256];
    __shared__ float Sig_s[16 * 17], L[16 * 17], Sf[16 * 17], A1[16 * 17];
    __shared__ float M[16 * 17], JT[16 * 17], D[16 * 17], T2[16 * 17];
    __shared__ float mu_s[16], mf[16], mp1[16], dm[16], wsh[3];

    for (int i = lane; i < KK * 256; i += 32) Ab[i] = Abase_g[i];
    if (lane < 16) mu_s[lane] = out_mu[((size_t)(TT - 1) * BB + b) * 16 + lane];
    for (int i = lane; i < 256; i += 32)
        Sig_s[(i >> 4) * 17 + (i & 15)] = out_Sig[((size_t)(TT - 1) * BB + b) * 256 + i];
    __syncthreads();

    for (int t = TT - 2; t >= 0; --t) {
        if (lane < 3) wsh[lane] = w_g[(b * TT + t + 1) * 3 + lane];
        if (lane < 16) {
            mf[lane] = out_mu[((size_t)t * BB + b) * 16 + lane];
            mp1[lane] = mu_pred[(size_t)(b * TT + t + 1) * 16 + lane];
        }
        __syncthreads();
        float w0 = wsh[0], w1 = wsh[1], w2 = wsh[2];
        for (int i = lane; i < 256; i += 32) {
            int rr = i >> 4, cc = i & 15;
            A1[rr * 17 + cc] = w0 * Ab[i] + w1 * Ab[256 + i] + w2 * Ab[512 + i];
            float sp = Sig_pred[(size_t)(b * TT + t + 1) * 256 + i];
            L[rr * 17 + cc] = sp;
            Sf[rr * 17 + cc] = out_Sig[((size_t)t * BB + b) * 256 + i];
            D[rr * 17 + cc] = Sig_s[rr * 17 + cc] - sp;
        }
        __syncthreads();

        // M = A1 @ Sf  (Sf symmetric => (Sf A1^T)^T = A1 Sf)
        store_c(M, 17, wmma_mm<16, false, false>(A1, 17, Sf, 17, 1.f, zero8()));
        __syncthreads();

        chol_n(L, 17, 16);
        chol_solve(L, 17, M, 17, JT, 17, 16);  // J^T = Sp1^{-1} A1 Sf

        if (lane < 16) dm[lane] = mu_s[lane] - mp1[lane];
        __syncthreads();
        if (lane < 16) {
            float s = mf[lane];
            for (int k = 0; k < 16; ++k) s += JT[k * 17 + lane] * dm[k];
            mu_s[lane] = s;
            out_mu[((size_t)t * BB + b) * 16 + lane] = s;
        }

        // Sig_new = Sf + (J D) J^T   (A frag transposed from JT; B = JT directly)
        store_c(T2, 17, wmma_mm<16, true, false>(JT, 17, D, 17, 1.f, zero8()));
        __syncthreads();
        v8f sf8 = load_c(Sf, 17);
        v8f sn = wmma_mm<16, false, false>(T2, 17, JT, 17, 1.f, sf8);
        store_c(Sig_s, 17, sn);
        __syncthreads();
        for (int i = lane; i < 256; i += 32)
            out_Sig[((size_t)t * BB + b) * 256 + i] = Sig_s[(i >> 4) * 17 + (i & 15)];
        __syncthreads();
    }
}

extern "C" void kernel_launch(void* const* d_in, const int* in_sizes, int n_in,
                              void* d_out, int out_size, void* d_ws, size_t ws_size,
                              hipStream_t stream) {
    const float* obs        = (const float*)d_in[0];
    const float* start_code = (const float*)d_in[1];
    const float* W1         = (const float*)d_in[2];
    const float* b1         = (const float*)d_in[3];
    const float* W2         = (const float*)d_in[4];
    const float* b2         = (const float*)d_in[5];
    const float* A          = (const float*)d_in[6];
    const float* C          = (const float*)d_in[7];

    float* w_ws     = (float*)d_ws;                              // B*T*3
    float* mu_pred  = w_ws + (size_t)BB * TT * 3;                // B*T*16
    float* Sig_pred = mu_pred + (size_t)BB * TT * 16;            // B*T*256

    float* out_mu  = (float*)d_out;                              // (T,B,16)
    float* out_Sig = out_mu + (size_t)TT * BB * 16;              // (T,B,16,16)

    kweights<<<(BB * TT) / 256, 256, 0, stream>>>(obs, start_code, W1, b1, W2, b2, w_ws);
    kfilter<<<BB, 32, 0, stream>>>(obs, A, C, w_ws, mu_pred, Sig_pred, out_mu, out_Sig);
    ksmooth<<<BB, 32, 0, stream>>>(A, w_ws, mu_pred, Sig_pred, out_mu, out_Sig);
}